// BMSampling_66683662238003
// MI455X (gfx1250) — compile-verified
//
#include <hip/hip_runtime.h>

// ---------------------------------------------------------------------------
// BMSampling as a skinny-K f32 GEMM on CDNA5 WMMA:
//   C[512, 320000] = A[512,100] (X) x B[100,320000] (smp_weight)
// ---------------------------------------------------------------------------

typedef float v2f __attribute__((ext_vector_type(2)));
typedef float v4f __attribute__((ext_vector_type(4)));
typedef float v8f __attribute__((ext_vector_type(8)));

#define M_DIM   512        // BATCH * CH
#define K_DIM   100        // T
#define N_DIM   320000     // N_SMP * D_PROP * T
#define WAVES   8          // waves per workgroup (wave32)
#define TPB     (WAVES * 32)
#define KSTEPS  (K_DIM / 4)   // 25 WMMA k-steps of 4
#define MTILES  (M_DIM / 16)  // 32
#define NTILES  (N_DIM / 16)  // 20000
#define LDS_BYTES (M_DIM * K_DIM * 4)  // 204800 B of the 320KB WGP LDS

static_assert(LDS_BYTES <= 320 * 1024, "fits WGP LDS");
static_assert(NTILES % WAVES == 0, "grid tiling exact");

__global__ __launch_bounds__(TPB, 1) void bm_gemm_wmma_f32(
    const float* __restrict__ X,    // [512,100] row-major
    const float* __restrict__ W,    // [100,320000] row-major
    float* __restrict__ out)        // [512,320000] row-major
{
    extern __shared__ float ldsA[];   // full A matrix, staged once per WG

    const int tid = threadIdx.x;

    // ---- Cooperative load of all of A into LDS (float4, fully coalesced) ----
    {
        const v4f* __restrict__ src = reinterpret_cast<const v4f*>(X);
        v4f* dst = reinterpret_cast<v4f*>(ldsA);
        const int n4 = (M_DIM * K_DIM) / 4;   // 12800
        #pragma unroll 4
        for (int i = tid; i < n4; i += TPB) dst[i] = src[i];
    }
    __syncthreads();

    const int lane   = tid & 31;
    const int waveid = tid >> 5;
    const int khalf  = lane >> 4;     // 0: K pair {0,1}; 1: K pair {2,3}
    const int nsub   = lane & 15;     // N (for B/C) or M (for A) within tile
    const int ntile  = blockIdx.x * WAVES + waveid;
    const long ncol  = (long)ntile * 16 + nsub;

    // ---- Load this wave's B column fragment for all of K (50 VGPRs) ----
    // B 4x16 layout: VGPR0 = row k0+2*khalf, VGPR1 = row k0+2*khalf+1,
    // lanes (mod 16) = N column. Each 64B line of W is read by exactly one
    // wave -> nontemporal keeps L2 for reuse-bearing data.
    v2f bfrag[KSTEPS];
    #pragma unroll
    for (int kk = 0; kk < KSTEPS; ++kk) {
        const int kr = 4 * kk + 2 * khalf;
        bfrag[kk].x = __builtin_nontemporal_load(&W[(long)(kr    ) * N_DIM + ncol]);
        bfrag[kk].y = __builtin_nontemporal_load(&W[(long)(kr + 1) * N_DIM + ncol]);
    }

    // ---- Sweep all M tiles, reusing the B fragment from registers ----
    for (int mt = 0; mt < MTILES; ++mt) {
        v8f acc = {};
        const int mrow = mt * 16 + nsub;   // A 16x4 layout: lane&15 = M row
        #pragma unroll
        for (int kk = 0; kk < KSTEPS; ++kk) {
            const int kr = 4 * kk + 2 * khalf;
            // conflict-free ds_load_b64: bank stride 36 mod 64 over 16 lanes
            v2f a = *reinterpret_cast<const v2f*>(&ldsA[mrow * K_DIM + kr]);
            acc = __builtin_amdgcn_wmma_f32_16x16x4_f32(
                /*neg_a=*/false, a, /*neg_b=*/false, bfrag[kk],
                /*c_mod=*/(short)0, acc, /*reuse_a=*/false, /*reuse_b=*/false);
        }
        // D 16x16 layout: VGPR r -> row r (lanes 0-15) / row r+8 (lanes 16-31)
        float* orow = out + (long)(mt * 16 + 8 * khalf) * N_DIM + ncol;
        #pragma unroll
        for (int r = 0; r < 8; ++r) {
            // 655MB streaming output: nontemporal so it never pollutes L2
            __builtin_nontemporal_store(acc[r], orow + (long)r * N_DIM);
        }
    }
}

extern "C" void kernel_launch(void* const* d_in, const int* in_sizes, int n_in,
                              void* d_out, int out_size, void* d_ws, size_t ws_size,
                              hipStream_t stream) {
    (void)in_sizes; (void)n_in; (void)out_size; (void)d_ws; (void)ws_size;

    const float* X = (const float*)d_in[0];   // [4,128,100] f32 == [512,100]
    const float* W = (const float*)d_in[1];   // [100,320000] f32
    float* out = (float*)d_out;               // [512,320000] f32

    // 204.8 KB dynamic LDS (> default static cap, <= 320KB WGP LDS)
    hipFuncSetAttribute((const void*)bm_gemm_wmma_f32,
                        hipFuncAttributeMaxDynamicSharedMemorySize, LDS_BYTES);

    dim3 grid(NTILES / WAVES);   // 2500 workgroups x 8 waves = 20000 N tiles
    bm_gemm_wmma_f32<<<grid, TPB, LDS_BYTES, stream>>>(X, W, out);
}